// ASH2DSelfAttention_64957085384891
// MI455X (gfx1250) — compile-verified
//
#include <hip/hip_runtime.h>
#include <math.h>

// Problem constants (from reference)
#define B_   2
#define T_   512
#define E_   128
#define H_   8
#define K_   8
#define VS_  64          // K*(4+GADD+RADD)
#define HID_ 512         // 4*E
#define BT_  (B_*T_)     // 1024
#define HE_  (H_*E_)     // 1024
#define BH_  (B_*H_)     // 16
#define NENT (BH_*T_*VS_) // 524288 sparse attention entries
#define KC_  64          // K-chunk staged into LDS per iteration

typedef __attribute__((ext_vector_type(2))) float v2f;
typedef __attribute__((ext_vector_type(8))) float v8f;

// ---------------- threefry2x32 (exact JAX PRNG core) ----------------
__host__ __device__ inline void threefry2x32(unsigned k0, unsigned k1,
                                             unsigned c0, unsigned c1,
                                             unsigned &o0, unsigned &o1) {
  unsigned ks2 = k0 ^ k1 ^ 0x1BD11BDAu;
  unsigned x0 = c0 + k0, x1 = c1 + k1;
#define TFR(r) { x0 += x1; x1 = (x1 << (r)) | (x1 >> (32 - (r))); x1 ^= x0; }
  TFR(13) TFR(15) TFR(26) TFR(6)
  x0 += k1;  x1 += ks2 + 1u;
  TFR(17) TFR(29) TFR(16) TFR(24)
  x0 += ks2; x1 += k0 + 2u;
  TFR(13) TFR(15) TFR(26) TFR(6)
  x0 += k0;  x1 += k1 + 3u;
  TFR(17) TFR(29) TFR(16) TFR(24)
  x0 += k1;  x1 += ks2 + 4u;
  TFR(13) TFR(15) TFR(26) TFR(6)
  x0 += ks2; x1 += k0 + 5u;
#undef TFR
  o0 = x0; o1 = x1;
}

// ordered-uint encoding for float atomicMax
__device__ inline unsigned f2o(float f) {
  unsigned u = __float_as_uint(f);
  return (u & 0x80000000u) ? ~u : (u | 0x80000000u);
}
__device__ inline float o2f(unsigned u) {
  unsigned v = (u & 0x80000000u) ? (u & 0x7FFFFFFFu) : ~u;
  return __uint_as_float(v);
}
#define ORDERED_NEG_INF 0x007FFFFFu

// async global->LDS staging helpers (CDNA5 GLOBAL_LOAD_ASYNC_TO_LDS_B128,
// tracked with ASYNCcnt; LDS offset = low 32 bits of flat shared pointer)
__device__ inline void ash_async_b128(const float* gsrc, float* ldst) {
  unsigned loff = (unsigned)(size_t)ldst;
  unsigned long long ga = (unsigned long long)(size_t)gsrc;
  asm volatile("global_load_async_to_lds_b128 %0, %1, off"
               :: "v"(loff), "v"(ga) : "memory");
}
__device__ inline void ash_wait_async0() {
#if __has_builtin(__builtin_amdgcn_s_wait_asynccnt)
  __builtin_amdgcn_s_wait_asynccnt(0);
#else
  asm volatile("s_wait_asynccnt 0x0" ::: "memory");
#endif
}

// ---------------- generic fp32 WMMA GEMM: C = A@B (+bias, optional relu) ---
// One wave (32 lanes) per 16x16 output tile; K stepped by 4 via
// V_WMMA_F32_16X16X4_F32. The 16-column B panel is staged per 64-row K-chunk
// into LDS with async B128 copies (8 coalesced 16B transfers replace 32
// strided b32 loads), then consumed via ds_load. M,N multiples of 16; K
// multiple of 4; B rows 16B-aligned.
__global__ __launch_bounds__(32)
void ash_wmma_gemm_f32(const float* __restrict__ A, const float* __restrict__ Bm,
                       const float* __restrict__ bias, float* __restrict__ C,
                       int M, int N, int K, int lda, int ldb, int ldc, int doRelu)
{
  __shared__ __align__(16) float Bs[KC_ * 16];
  int ntn  = N >> 4;
  int tile = blockIdx.x;
  int m0 = (tile / ntn) << 4;
  int n0 = (tile % ntn) << 4;
  int lane = threadIdx.x;
  int half = lane >> 4;     // 0: K rows {0,1}; 1: K rows {2,3} within step
  int l16  = lane & 15;
  int rcp  = lane >> 2;        // copy role: row within 8-row group
  int ccp  = (lane & 3) * 4;   // copy role: 4-float column chunk

  v8f acc = {};
  const float* aptr = A + (size_t)(m0 + l16) * lda;

  int kk = 0;
  for (; kk + KC_ <= K; kk += KC_) {
    // stage B[kk..kk+KC_) x [n0..n0+16) -> LDS (async, ASYNCcnt-tracked)
#pragma unroll
    for (int it = 0; it < KC_ / 8; ++it) {
      int r = it * 8 + rcp;
      ash_async_b128(Bm + (size_t)(kk + r) * ldb + n0 + ccp, &Bs[r * 16 + ccp]);
    }
    ash_wait_async0();   // single wave owns this LDS tile; no barrier needed
#pragma unroll
    for (int k4 = 0; k4 < KC_; k4 += 4) {
      int kb = k4 + half * 2;
      v2f av, bv;
      av.x = aptr[kk + kb];
      av.y = aptr[kk + kb + 1];
      bv.x = Bs[kb * 16 + l16];
      bv.y = Bs[(kb + 1) * 16 + l16];
      acc = __builtin_amdgcn_wmma_f32_16x16x4_f32(
          false, av, false, bv, (short)0, acc, false, false);
    }
    // all ds_loads are consumed by the wmmas above (compiler waits DScnt),
    // so re-staging next chunk cannot race the reads
  }
  for (; kk < K; kk += 4) {          // K remainder (e.g. K=132): direct path
    int kb = kk + half * 2;
    v2f av, bv;
    av.x = aptr[kb];
    av.y = aptr[kb + 1];
    bv.x = Bm[(size_t)kb       * ldb + n0 + l16];
    bv.y = Bm[(size_t)(kb + 1) * ldb + n0 + l16];
    acc = __builtin_amdgcn_wmma_f32_16x16x4_f32(
        false, av, false, bv, (short)0, acc, false, false);
  }

  float bb = bias ? bias[n0 + l16] : 0.0f;
#pragma unroll
  for (int i = 0; i < 8; ++i) {
    float v = acc[i] + bb;
    if (doRelu) v = fmaxf(v, 0.0f);
    C[(size_t)(m0 + i + half * 8) * ldc + n0 + l16] = v;
  }
}

// ---------------- prep kernels (padding / concat) ----------------
__global__ void ash_prep_xc(const float* __restrict__ x, float* __restrict__ Xc) {
  int i = blockIdx.x * blockDim.x + threadIdx.x;      // BT_*132
  if (i >= BT_ * 132) return;
  int r = i / 132, c = i % 132;
  float v;
  if (c < E_)       v = x[(size_t)r * E_ + c];
  else if (c == E_) v = (float)(r % T_) / (float)T_;  // coords
  else              v = 0.0f;
  Xc[i] = v;
}
__global__ void ash_prep_w1(const float* __restrict__ W1, float* __restrict__ W1p) {
  int i = blockIdx.x * blockDim.x + threadIdx.x;      // 132*HID_
  if (i >= 132 * HID_) return;
  int r = i / HID_, c = i % HID_;
  W1p[i] = (r < E_ + 1) ? W1[(size_t)r * HID_ + c] : 0.0f;
}
__global__ void ash_prep_w2(const float* __restrict__ W2, const float* __restrict__ b2,
                            float* __restrict__ W2p, float* __restrict__ b2p) {
  int i = blockIdx.x * blockDim.x + threadIdx.x;      // HID_*32
  if (i < HID_ * 32) {
    int r = i / 32, c = i % 32;
    W2p[i] = (c < 24) ? W2[(size_t)r * 24 + c] : 0.0f;
  }
  if (i < 32) b2p[i] = (i < 24) ? b2[i] : 0.0f;
}
__global__ void ash_init(unsigned* __restrict__ mxo, float* __restrict__ denom,
                         float* __restrict__ acc) {
  int i = blockIdx.x * blockDim.x + threadIdx.x;
  if (i < BH_ * T_) { mxo[i] = ORDERED_NEG_INF; denom[i] = 0.0f; }
  if (i < BT_ * HE_) acc[i] = 0.0f;
}

// ---------------- params -> means/sigmas -> indices/weights ----------------
__global__ void ash_params_to_indices(const float* __restrict__ P,
                                      int* __restrict__ idxb, float* __restrict__ wb,
                                      unsigned rk1a, unsigned rk1b,
                                      unsigned rk2a, unsigned rk2b)
{
  int tid = blockIdx.x * blockDim.x + threadIdx.x;
  if (tid >= BT_) return;
  int bi = tid / T_, ti = tid % T_;
  const float* p = P + (size_t)tid * 32;

  float sc   = (float)ti / (float)T_ * 0.999f + 0.0005f;
  float diag = logf(sc / (1.0f - sc));

  float m0[K_], m1[K_], sg[K_];
#pragma unroll
  for (int k = 0; k < K_; ++k) {
    // means flipped before sigmoid: new0 uses p[2k+1], new1 uses p[2k] (MMULT=1)
    float a = diag + p[2 * k + 1];
    float b = diag + p[2 * k];
    m0[k] = (1.0f / (1.0f + expf(-a))) * (float)(T_ - 1);
    m1[k] = (1.0f / (1.0f + expf(-b))) * (float)(T_ - 1);
    float s  = p[16 + k] + 2.0f;                  // + SIGMA_BOOST
    float sp = (s > 20.0f) ? s : log1pf(expf(s)); // softplus
    sg[k] = (sp + 0.05f) * 0.1f;                  // (+MIN_SIGMA)*SIGMA_SCALE
  }

  int idx0[VS_], idx1[VS_];  // FINAL (flipped) pairs: idx0=row, idx1=col
  const unsigned HALF = (unsigned)(B_ * T_ * K_ * 2 * 2) / 2u; // 16384
  for (int k = 0; k < K_; ++k) {
    int fl0 = (int)floorf(m0[k]);
    int fl1 = (int)floorf(m1[k]);
    int base = k * 8;
    const int o0[4] = {0, 0, 1, 1};
    const int o1[4] = {0, 1, 0, 1};
#pragma unroll
    for (int o = 0; o < 4; ++o) {
      int c0 = min(max(fl0 + o0[o], 0), T_ - 1);
      int c1 = min(max(fl1 + o1[o], 0), T_ - 1);
      idx0[base + o] = c1; idx1[base + o] = c0;   // final flip
    }
    for (int g = 0; g < 2; ++g) {                  // glob = randint(rk1,..,0,512)
      int c[2];
      for (int d = 0; d < 2; ++d) {
        unsigned f = ((((unsigned)bi * T_ + (unsigned)ti) * K_ + (unsigned)k) * 2u
                      + (unsigned)g) * 2u + (unsigned)d;
        unsigned w0, w1;
        if (f < HALF) { threefry2x32(rk1a, rk1b, f, f + HALF, w0, w1); c[d] = (int)(w0 & 511u); }
        else          { threefry2x32(rk1a, rk1b, f - HALF, f, w0, w1); c[d] = (int)(w1 & 511u); }
      }
      idx0[base + 4 + g] = c[1]; idx1[base + 4 + g] = c[0];
    }
    int lo0 = min(max(fl0 - 8, 0), T_ - 16);
    int lo1 = min(max(fl1 - 8, 0), T_ - 16);
    for (int r = 0; r < 2; ++r) {                  // rel = lower + randint(rk2,..,0,16)
      int c[2];
      for (int d = 0; d < 2; ++d) {
        unsigned f = ((((unsigned)bi * T_ + (unsigned)ti) * K_ + (unsigned)k) * 2u
                      + (unsigned)r) * 2u + (unsigned)d;
        unsigned w0, w1;
        if (f < HALF) { threefry2x32(rk2a, rk2b, f, f + HALF, w0, w1); c[d] = (int)(w0 & 15u); }
        else          { threefry2x32(rk2a, rk2b, f - HALF, f, w0, w1); c[d] = (int)(w1 & 15u); }
      }
      idx0[base + 6 + r] = lo1 + c[1]; idx1[base + 6 + r] = lo0 + c[0];
    }
  }

  // duplicate suppression (strictly-lower-triangle equality)
  unsigned long long dupmask = 0ull;
  for (int i = 1; i < VS_; ++i) {
    bool d = false;
    for (int j = 0; j < i; ++j)
      d |= (idx0[i] == idx0[j]) & (idx1[i] == idx1[j]);
    if (d) dupmask |= (1ull << i);
  }

  // Gaussian props, normalized over vs per component k, summed over k
  float ksum[K_];
#pragma unroll
  for (int k = 0; k < K_; ++k) ksum[k] = 0.0f;
  for (int v = 0; v < VS_; ++v) {
    if ((dupmask >> v) & 1ull) continue;
    float f0 = (float)idx0[v], f1 = (float)idx1[v];
#pragma unroll
    for (int k = 0; k < K_; ++k) {
      float z0 = (f0 - m0[k]) / sg[k];
      float z1 = (f1 - m1[k]) / sg[k];
      ksum[k] += expf(-0.5f * (z0 * z0 + z1 * z1));
    }
  }
  for (int v = 0; v < VS_; ++v) {
    float w = 0.0f;
    if (!((dupmask >> v) & 1ull)) {
      float f0 = (float)idx0[v], f1 = (float)idx1[v];
#pragma unroll
      for (int k = 0; k < K_; ++k) {
        float z0 = (f0 - m0[k]) / sg[k];
        float z1 = (f1 - m1[k]) / sg[k];
        w += expf(-0.5f * (z0 * z0 + z1 * z1)) / ksum[k];
      }
    }
    wb[(size_t)tid * VS_ + v] = w;
    idxb[((size_t)tid * VS_ + v) * 2 + 0] = idx0[v];
    idxb[((size_t)tid * VS_ + v) * 2 + 1] = idx1[v];
  }
}

// ---------------- sparse attention passes ----------------
__device__ inline void ash_decode(int e, int &b, int &h, int &t, int &v) {
  v = e % VS_; int rem = e / VS_;
  t = rem % T_; int bh = rem / T_;
  b = bh / H_; h = bh % H_;
}

__global__ void ash_pass1(const float* __restrict__ Q, const float* __restrict__ Kb,
                          const int* __restrict__ idxb, const float* __restrict__ wb,
                          float* __restrict__ valb, unsigned* __restrict__ mxo)
{
  int e = blockIdx.x * blockDim.x + threadIdx.x;
  if (e >= NENT) return;
  int b, h, t, v; ash_decode(e, b, h, t, v);
  const int2 idx = ((const int2*)idxb)[(size_t)(b * T_ + t) * VS_ + v];
  int row = idx.x, col = idx.y;
  const float4* q = (const float4*)(Q  + (size_t)(b * T_ + row) * HE_ + h * E_);
  const float4* k = (const float4*)(Kb + (size_t)(b * T_ + col) * HE_ + h * E_);
  float dot = 0.0f;
#pragma unroll 8
  for (int d = 0; d < E_ / 4; ++d) {
    float4 qa = q[d], ka = k[d];
    dot += qa.x * ka.x + qa.y * ka.y + qa.z * ka.z + qa.w * ka.w;
  }
  float val = wb[(size_t)(b * T_ + t) * VS_ + v] * dot * (1.0f / sqrtf((float)E_));
  valb[e] = val;
  atomicMax(&mxo[(b * H_ + h) * T_ + row], f2o(val));
}

__global__ void ash_pass2(float* __restrict__ valb, const unsigned* __restrict__ mxo,
                          const int* __restrict__ idxb, float* __restrict__ denom)
{
  int e = blockIdx.x * blockDim.x + threadIdx.x;
  if (e >= NENT) return;
  int b, h, t, v; ash_decode(e, b, h, t, v);
  int row = idxb[((size_t)(b * T_ + t) * VS_ + v) * 2 + 0];
  int seg = (b * H_ + h) * T_ + row;
  float mxf = o2f(mxo[seg]);
  if ((__float_as_uint(mxf) & 0x7F800000u) == 0x7F800000u) mxf = 0.0f; // !isfinite
  float ex = expf(valb[e] - mxf);
  valb[e] = ex;
  atomicAdd(&denom[seg], ex);
}

// wave-per-entry: 32 lanes x 4 dims cover E_=128; exposes the scatter
// bandwidth (the only real HBM consumer in this problem) in parallel
__global__ void ash_pass3(const float* __restrict__ valb, const float* __restrict__ denom,
                          const int* __restrict__ idxb, const float* __restrict__ Vb,
                          float* __restrict__ acc)
{
  int gid  = blockIdx.x * blockDim.x + threadIdx.x;
  int e    = gid >> 5;
  int lane = gid & 31;
  if (e >= NENT) return;
  int b, h, t, v; ash_decode(e, b, h, t, v);
  const int2 idx = ((const int2*)idxb)[(size_t)(b * T_ + t) * VS_ + v];
  int row = idx.x, col = idx.y;
  int seg = (b * H_ + h) * T_ + row;
  float att = valb[e] / denom[seg];
  const float4 vv = *(const float4*)(Vb + (size_t)(b * T_ + col) * HE_ + h * E_ + lane * 4);
  float* ap = acc + (size_t)(b * T_ + row) * HE_ + h * E_ + lane * 4;
  atomicAdd(&ap[0], att * vv.x);
  atomicAdd(&ap[1], att * vv.y);
  atomicAdd(&ap[2], att * vv.z);
  atomicAdd(&ap[3], att * vv.w);
}

// ---------------- host entry ----------------
extern "C" void kernel_launch(void* const* d_in, const int* in_sizes, int n_in,
                              void* d_out, int out_size, void* d_ws, size_t ws_size,
                              hipStream_t stream) {
  const float* x  = (const float*)d_in[0];
  const float* Wq = (const float*)d_in[1];
  const float* Wk = (const float*)d_in[2];
  const float* Wv = (const float*)d_in[3];
  const float* Wu = (const float*)d_in[4];
  const float* bu = (const float*)d_in[5];
  const float* W1 = (const float*)d_in[6];
  const float* b1 = (const float*)d_in[7];
  const float* W2 = (const float*)d_in[8];
  const float* b2 = (const float*)d_in[9];
  float* out = (float*)d_out;

  // bump allocator over workspace
  char* wsb = (char*)d_ws;
  size_t off = 0;
  auto alloc = [&](size_t bytes) -> void* {
    off = (off + 255) & ~(size_t)255;
    void* p = wsb + off;
    off += bytes;
    return p;
  };
  float* Xc    = (float*)alloc((size_t)BT_ * 132 * 4);
  float* W1p   = (float*)alloc((size_t)132 * HID_ * 4);
  float* Hbuf  = (float*)alloc((size_t)BT_ * HID_ * 4);
  float* W2p   = (float*)alloc((size_t)HID_ * 32 * 4);
  float* b2p   = (float*)alloc(32 * 4);
  float* Pbuf  = (float*)alloc((size_t)BT_ * 32 * 4);
  int*   idxb  = (int*)  alloc((size_t)BT_ * VS_ * 2 * 4);
  float* wbuf  = (float*)alloc((size_t)BT_ * VS_ * 4);
  float* Qb    = (float*)alloc((size_t)BT_ * HE_ * 4);
  float* Kbf   = (float*)alloc((size_t)BT_ * HE_ * 4);
  float* Vb    = (float*)alloc((size_t)BT_ * HE_ * 4);
  float* valb  = (float*)alloc((size_t)NENT * 4);
  unsigned* mxo = (unsigned*)alloc((size_t)BH_ * T_ * 4);
  float* denom = (float*)alloc((size_t)BH_ * T_ * 4);
  float* accb  = (float*)alloc((size_t)BT_ * HE_ * 4);
  (void)ws_size; (void)n_in; (void)in_sizes; (void)out_size;

  // JAX key derivation on host: split(key(42)) via threefry2x32
  unsigned rk1a, rk1b, rk2a, rk2b;
  threefry2x32(0u, 42u, 0u, 2u, rk1a, rk2a);
  threefry2x32(0u, 42u, 1u, 3u, rk1b, rk2b);

  // init + prep
  ash_init<<<(BT_ * HE_ + 255) / 256, 256, 0, stream>>>(mxo, denom, accb);
  ash_prep_xc<<<(BT_ * 132 + 255) / 256, 256, 0, stream>>>(x, Xc);
  ash_prep_w1<<<(132 * HID_ + 255) / 256, 256, 0, stream>>>(W1, W1p);
  ash_prep_w2<<<(HID_ * 32 + 255) / 256, 256, 0, stream>>>(W2, b2, W2p, b2p);

  // hyper-MLP: H = relu(Xc @ W1p + b1); P = H @ W2p + b2p
  ash_wmma_gemm_f32<<<(BT_ / 16) * (HID_ / 16), 32, 0, stream>>>(
      Xc, W1p, b1, Hbuf, BT_, HID_, 132, 132, HID_, HID_, 1);
  ash_wmma_gemm_f32<<<(BT_ / 16) * (32 / 16), 32, 0, stream>>>(
      Hbuf, W2p, b2p, Pbuf, BT_, 32, HID_, HID_, 32, 32, 0);

  // indices + weights
  ash_params_to_indices<<<(BT_ + 255) / 256, 256, 0, stream>>>(
      Pbuf, idxb, wbuf, rk1a, rk1b, rk2a, rk2b);

  // Q/K/V projections (scale folded into pass1)
  ash_wmma_gemm_f32<<<(BT_ / 16) * (HE_ / 16), 32, 0, stream>>>(
      x, Wq, nullptr, Qb, BT_, HE_, E_, E_, HE_, HE_, 0);
  ash_wmma_gemm_f32<<<(BT_ / 16) * (HE_ / 16), 32, 0, stream>>>(
      x, Wk, nullptr, Kbf, BT_, HE_, E_, E_, HE_, HE_, 0);
  ash_wmma_gemm_f32<<<(BT_ / 16) * (HE_ / 16), 32, 0, stream>>>(
      x, Wv, nullptr, Vb, BT_, HE_, E_, E_, HE_, HE_, 0);

  // sparse scatter-softmax attention
  ash_pass1<<<(NENT + 255) / 256, 256, 0, stream>>>(Qb, Kbf, idxb, wbuf, valb, mxo);
  ash_pass2<<<(NENT + 255) / 256, 256, 0, stream>>>(valb, mxo, idxb, denom);
  ash_pass3<<<((NENT * 32) + 255) / 256, 256, 0, stream>>>(valb, denom, idxb, Vb, accb);

  // output projection: out = acc @ Wu + bu
  ash_wmma_gemm_f32<<<(BT_ / 16) * (E_ / 16), 32, 0, stream>>>(
      accb, Wu, bu, out, BT_, E_, HE_, HE_, E_, E_, 0);
}